// WaveletBlock1D_70669391888565
// MI455X (gfx1250) — compile-verified
//
#include <hip/hip_runtime.h>
#include <math.h>

typedef __attribute__((ext_vector_type(2))) float v2f;
typedef __attribute__((ext_vector_type(8))) float v8f;
typedef __attribute__((ext_vector_type(4))) unsigned int v4u;
typedef __attribute__((ext_vector_type(8))) int v8i;
typedef __attribute__((ext_vector_type(4))) int v4i;

#define B_   16
#define N_   8192
#define C_   128
#define NC   1024                     // N >> 3
#define INV_2SQRT2 0.35355339059327373f

__device__ __forceinline__ float gelu_tanh(float t) {
    float u = 0.7978845608028654f * (t + 0.044715f * t * t * t);
    return 0.5f * t * (1.0f + tanhf(u));
}

// ---------------------------------------------------------------------------
// TDM: 1-D contiguous copy (nelem f32) global -> LDS via TENSOR_LOAD_TO_LDS.
// D# per CDNA5 ISA:
//   group0: count=1 | lds_addr[63:32] | global_addr[120:64] | type=2[127:126]
//   group1: data_size=4B[17:16], tensor_dim0=nelem[79:48], tensor_dim1=1[111:80],
//           tile_dim0=nelem[127:112], tile_dim1=1[143:128], dim0_stride=nelem[207:160]
// Groups 2/3 zero (<=2D tensor). This toolchain uses the 6-arg builtin form
// (extra int32x8 group before cpol). Issue once per workgroup (wave 0); TDM
// ignores EXEC; TENSORcnt is per-wave so the issuing wave waits, then barrier.
// ---------------------------------------------------------------------------
__device__ __forceinline__ void tdm_load_1d(void* lds_ptr, const void* gptr,
                                            unsigned int nelem) {
    unsigned long long ga = (unsigned long long)(uintptr_t)gptr;
    unsigned int lds_addr = (unsigned int)(uintptr_t)lds_ptr;   // low 32 = LDS byte offset
    v4u g0;
    g0.x = 1u;                                   // count=1 (user descriptor)
    g0.y = lds_addr;                             // lds_addr (bytes)
    g0.z = (unsigned int)ga;                     // global_addr[31:0]
    g0.w = (unsigned int)(ga >> 32) | 0x80000000u; // global_addr[56:32] | type=2
    v8i g1;
    g1[0] = 0x00020000;                          // workgroup_mask=0, data_size=2 (4B)
    g1[1] = (int)((nelem & 0xFFFFu) << 16);      // tensor_dim0 lo16 -> bits 63:48
    g1[2] = (int)((nelem >> 16) | (1u << 16));   // tensor_dim0 hi16, tensor_dim1=1
    g1[3] = (int)((nelem & 0xFFFFu) << 16);      // tile_dim0 -> bits 127:112
    g1[4] = 1;                                   // tile_dim1=1, tile_dim2=0
    g1[5] = (int)nelem;                          // tensor_dim0_stride lo32
    g1[6] = 0;                                   // stride hi / dim1_stride lo
    g1[7] = 0;
    v4i z4 = {0, 0, 0, 0};
    v8i z8 = {0, 0, 0, 0, 0, 0, 0, 0};
    __builtin_amdgcn_tensor_load_to_lds(g0, g1, z4, z4, z8, 0);
}

// ---------------------------------------------------------------------------
// Phase 1: coarse Haar bands. a3/d3 stored as (k, b, i) so the WMMA A-fragment
// load in phase 3 is a per-lane 8-byte contiguous load.
// ---------------------------------------------------------------------------
__global__ __launch_bounds__(256)
void k_coarse(const float* __restrict__ x,
              float* __restrict__ a3, float* __restrict__ d3) {
    int gid = blockIdx.x * 256 + threadIdx.x;          // 16*1024*128 threads
    int c = gid & (C_ - 1);
    int k = (gid >> 7) & (NC - 1);
    int b = gid >> 17;
    const float* p = x + ((size_t)(b * N_ + k * 8) * C_) + c;
    float s0 = 0.f, s1 = 0.f;
#pragma unroll
    for (int t = 0; t < 4; ++t) s0 += p[t * C_];
#pragma unroll
    for (int t = 4; t < 8; ++t) s1 += p[t * C_];
    int o = (k * B_ + b) * C_ + c;
    a3[o] = (s0 + s1) * INV_2SQRT2;
    d3[o] = (s0 - s1) * INV_2SQRT2;
}

// ---------------------------------------------------------------------------
// Phase 2: tiled transpose w[(i*128+o)][k] -> wt[k][(i*128+o)] so phase 3 can
// stream 64 KiB contiguous per-k slices. Classic 32x33 LDS tile, fully
// coalesced both directions.
// ---------------------------------------------------------------------------
__global__ void k_transpose(const float* __restrict__ w1, const float* __restrict__ w2,
                            float* __restrict__ w1t, float* __restrict__ w2t) {
    __shared__ float tile[32][33];
    const float* src = blockIdx.z ? w2 : w1;
    float* dst       = blockIdx.z ? w2t : w1t;
    int k0 = blockIdx.x * 32;                 // source col block (k, contiguous)
    int r0 = blockIdx.y * 32;                 // source row block (r = i*128+o)
    int tx = threadIdx.x, ty = threadIdx.y;   // 32 x 8
#pragma unroll
    for (int j = 0; j < 32; j += 8)
        tile[ty + j][tx] = src[(size_t)(r0 + ty + j) * NC + k0 + tx];
    __syncthreads();
#pragma unroll
    for (int j = 0; j < 32; j += 8)
        dst[(size_t)(k0 + ty + j) * (C_ * C_) + r0 + tx] = tile[tx][ty + j];
}

// ---------------------------------------------------------------------------
// Phase 3: per-k einsum, D = a3(16x128) * w[:,:,k](128x128) via f32 WMMA.
// One workgroup per k; TDM stages the two 64 KiB w-slices into 128 KiB LDS;
// 8 waves each own one 16-wide o-tile and accumulate both products.
// Writes dA/dD = (A - a3), (D - d3) in (b,k,c) layout.
// ---------------------------------------------------------------------------
__global__ __launch_bounds__(256)
void k_einsum(const float* __restrict__ a3, const float* __restrict__ d3,
              const float* __restrict__ w1t, const float* __restrict__ w2t,
              float* __restrict__ dA, float* __restrict__ dD) {
    extern __shared__ float smem[];           // 2 * 16384 floats = 128 KiB
    float* wlA = smem;
    float* wlD = smem + C_ * C_;
    int k = blockIdx.x;
    int tid = threadIdx.x;

    if (tid < 32) {                           // wave 0 drives the TDM
        tdm_load_1d(wlA, w1t + (size_t)k * C_ * C_, C_ * C_);
        tdm_load_1d(wlD, w2t + (size_t)k * C_ * C_, C_ * C_);
        __builtin_amdgcn_s_wait_tensorcnt(0);
    }
    __syncthreads();

    int wid = tid >> 5, lane = tid & 31;
    int half = lane >> 4, lm = lane & 15;
    int o0 = wid * 16;
    const float* a3k = a3 + (size_t)k * B_ * C_;   // (b, i)
    const float* d3k = d3 + (size_t)k * B_ * C_;

    v8f accA{}; v8f accD{};
#pragma unroll 4
    for (int s = 0; s < 32; ++s) {
        int i0 = s * 4 + 2 * half;            // A-frag: lanes 0-15 K=0,1 ; 16-31 K=2,3
        v2f aA = *(const v2f*)(a3k + lm * C_ + i0);
        v2f aD = *(const v2f*)(d3k + lm * C_ + i0);
        v2f bA, bD;                           // B-frag: row K, col o0+lm
        bA.x = wlA[i0 * C_ + o0 + lm];  bA.y = wlA[(i0 + 1) * C_ + o0 + lm];
        bD.x = wlD[i0 * C_ + o0 + lm];  bD.y = wlD[(i0 + 1) * C_ + o0 + lm];
        accA = __builtin_amdgcn_wmma_f32_16x16x4_f32(false, aA, false, bA,
                                                     (short)0, accA, false, false);
        accD = __builtin_amdgcn_wmma_f32_16x16x4_f32(false, aD, false, bD,
                                                     (short)0, accD, false, false);
    }

#pragma unroll
    for (int r = 0; r < 8; ++r) {             // C/D layout: M = r + 8*half, N = lm
        int b = r + 8 * half;
        int c = o0 + lm;
        size_t oidx = ((size_t)b * NC + k) * C_ + c;
        dA[oidx] = accA[r] - a3k[b * C_ + c];
        dD[oidx] = accD[r] - d3k[b * C_ + c];
    }
}

// ---------------------------------------------------------------------------
// Phase 4: shortcut GEMM (131072x128x128) fused with the wavelet spread and
// tanh-GELU. 64 rows / workgroup; TDM stages dense_w (64K) + x-block (32K)
// into LDS; 8 waves x 4 M-tiles, 32 wmma per tile.
// ---------------------------------------------------------------------------
__global__ __launch_bounds__(256)
void k_dense(const float* __restrict__ x, const float* __restrict__ dw,
             const float* __restrict__ db, const float* __restrict__ dA,
             const float* __restrict__ dD, float* __restrict__ out) {
    extern __shared__ float smem[];
    float* wd = smem;                 // 16384 floats
    float* xs = smem + C_ * C_;       // 8192 floats
    float* bs = xs + 64 * C_;         // 128 floats
    int tid = threadIdx.x;
    size_t row0 = (size_t)blockIdx.x * 64;

    if (tid < 32) {                   // wave 0 drives the TDM
        tdm_load_1d(wd, dw, C_ * C_);
        tdm_load_1d(xs, x + row0 * C_, 64 * C_);
        __builtin_amdgcn_s_wait_tensorcnt(0);
    } else if (tid >= 32 && tid < 32 + C_) {
        bs[tid - 32] = db[tid - 32];
    }
    __syncthreads();

    int wid = tid >> 5, lane = tid & 31;
    int half = lane >> 4, lm = lane & 15;
    int c0 = wid * 16;

    for (int mt = 0; mt < 4; ++mt) {
        v8f acc{};
#pragma unroll 4
        for (int s = 0; s < 32; ++s) {
            int i0 = s * 4 + 2 * half;
            v2f a = *(const v2f*)(xs + (mt * 16 + lm) * C_ + i0);
            v2f b;
            b.x = wd[i0 * C_ + c0 + lm];
            b.y = wd[(i0 + 1) * C_ + c0 + lm];
            acc = __builtin_amdgcn_wmma_f32_16x16x4_f32(false, a, false, b,
                                                        (short)0, acc, false, false);
        }
        int c = c0 + lm;
#pragma unroll
        for (int r = 0; r < 8; ++r) {
            int lrow = mt * 16 + r + 8 * half;
            size_t R = row0 + lrow;           // global row = b*8192 + n
            int n = (int)(R & (N_ - 1));
            int bidx = (int)(R >> 13);
            int kk = n >> 3;
            float sgn = ((n & 7) < 4) ? 1.0f : -1.0f;
            size_t didx = ((size_t)bidx * NC + kk) * C_ + c;
            float t = acc[r] + bs[c] + xs[lrow * C_ + c]
                    + (dA[didx] + sgn * dD[didx]) * INV_2SQRT2;
            out[R * C_ + c] = gelu_tanh(t);
        }
    }
}

// ---------------------------------------------------------------------------
extern "C" void kernel_launch(void* const* d_in, const int* in_sizes, int n_in,
                              void* d_out, int out_size, void* d_ws, size_t ws_size,
                              hipStream_t stream) {
    const float* x  = (const float*)d_in[0];
    const float* w1 = (const float*)d_in[1];
    const float* w2 = (const float*)d_in[2];
    const float* dw = (const float*)d_in[3];
    const float* db = (const float*)d_in[4];
    float* out = (float*)d_out;

    // workspace layout (floats): a3, d3, dA, dD (2M each) + w1t, w2t (16M each)
    float* a3  = (float*)d_ws;
    float* d3  = a3 + 2097152;
    float* dA  = d3 + 2097152;
    float* dD  = dA + 2097152;
    float* w1t = dD + 2097152;
    float* w2t = w1t + 16777216;

    k_coarse<<<8192, 256, 0, stream>>>(x, a3, d3);

    dim3 tg(NC / 32, (C_ * C_) / 32, 2), tb(32, 8, 1);
    k_transpose<<<tg, tb, 0, stream>>>(w1, w2, w1t, w2t);

    k_einsum<<<NC, 256, 2 * C_ * C_ * sizeof(float), stream>>>(a3, d3, w1t, w2t, dA, dD);

    size_t dyn = (C_ * C_ + 64 * C_ + C_) * sizeof(float);
    k_dense<<<(B_ * N_) / 64, 256, dyn, stream>>>(x, dw, db, dA, dD, out);
}